// GraphSAGE_51419348468006
// MI455X (gfx1250) — compile-verified
//
#include <hip/hip_runtime.h>

typedef __attribute__((ext_vector_type(2))) float v2f;
typedef __attribute__((ext_vector_type(8))) float v8f;

#define N_NODES 100000
#define N_EDGES 1600000
#define IN_CH 32
#define HID_CH 64

// ---------------------------------------------------------------- zero scratch
__global__ void sage_zero_kernel(float* __restrict__ p, long n) {
    long i = (long)blockIdx.x * blockDim.x + threadIdx.x;
    if (i < n) p[i] = 0.0f;
}

// ------------------------------------------------ layer-1 edge scatter (32 ch)
// thread t -> edge e = t/32, channel c = t%32
__global__ void sage_agg1_kernel(const float* __restrict__ x,
                                 const long long* __restrict__ src,
                                 const long long* __restrict__ dst,
                                 float* __restrict__ agg1,
                                 float* __restrict__ deg) {
    long t = (long)blockIdx.x * blockDim.x + threadIdx.x;
    if (t >= (long)N_EDGES * IN_CH) return;
    int e = (int)(t >> 5);
    int c = (int)(t & 31);
    int s = (int)src[e];
    int d = (int)dst[e];
    atomicAdd(&agg1[(long)d * IN_CH + c], x[(long)s * IN_CH + c]);
    if (c == 0) atomicAdd(&deg[d], 1.0f);
}

// ---------------------------------------- layer-1 fused GEMM + bias + ReLU
// h[16x64 tile] = relu( [mean | x](16x64) @ [W1l ; W1r](64x64) + b1 )
// One wave per 16-node tile; fp32 WMMA 16x16x4, 16 K-steps x 4 N-tiles.
__global__ void sage_l1_wmma_kernel(const float* __restrict__ agg1,
                                    const float* __restrict__ deg,
                                    const float* __restrict__ x,
                                    const float* __restrict__ W1l,  // [32][64]
                                    const float* __restrict__ W1r,  // [32][64]
                                    const float* __restrict__ b1,   // [64]
                                    float* __restrict__ h) {
    const int lane = threadIdx.x & 31;
    const int wave = threadIdx.x >> 5;
    const int tile = blockIdx.x * 8 + wave;          // 16-node tiles
    if (tile >= N_NODES / 16) return;                // wave-uniform guard

    const int n15  = lane & 15;
    const int koff = (lane < 16) ? 0 : 2;            // ISA A/B layout: lane half -> K offset
    const int row  = tile * 16 + n15;                // M = lane % 16

    const float inv = 1.0f / fmaxf(deg[row], 1.0f);

    const float* __restrict__ arow = agg1 + (long)row * IN_CH;
    const float* __restrict__ xrow = x    + (long)row * IN_CH;

    v8f acc0 = {}, acc1 = {}, acc2 = {}, acc3 = {};

    // K = 0..31 : mean part vs W1_l
#pragma unroll
    for (int kk = 0; kk < 8; ++kk) {
        const int k = kk * 4 + koff;
        v2f a;
        a.x = arow[k]     * inv;
        a.y = arow[k + 1] * inv;
        const float* __restrict__ Wk  = W1l + (long)k * HID_CH;
        const float* __restrict__ Wk1 = Wk + HID_CH;
        v2f b;
        b.x = Wk[n15];        b.y = Wk1[n15];
        acc0 = __builtin_amdgcn_wmma_f32_16x16x4_f32(false, a, false, b, (short)0, acc0, false, false);
        b.x = Wk[16 + n15];   b.y = Wk1[16 + n15];
        acc1 = __builtin_amdgcn_wmma_f32_16x16x4_f32(false, a, false, b, (short)0, acc1, false, false);
        b.x = Wk[32 + n15];   b.y = Wk1[32 + n15];
        acc2 = __builtin_amdgcn_wmma_f32_16x16x4_f32(false, a, false, b, (short)0, acc2, false, false);
        b.x = Wk[48 + n15];   b.y = Wk1[48 + n15];
        acc3 = __builtin_amdgcn_wmma_f32_16x16x4_f32(false, a, false, b, (short)0, acc3, false, false);
    }

    // K = 32..63 : self part vs W1_r
#pragma unroll
    for (int kk = 0; kk < 8; ++kk) {
        const int k = kk * 4 + koff;
        v2f a;
        a.x = xrow[k];
        a.y = xrow[k + 1];
        const float* __restrict__ Wk  = W1r + (long)k * HID_CH;
        const float* __restrict__ Wk1 = Wk + HID_CH;
        v2f b;
        b.x = Wk[n15];        b.y = Wk1[n15];
        acc0 = __builtin_amdgcn_wmma_f32_16x16x4_f32(false, a, false, b, (short)0, acc0, false, false);
        b.x = Wk[16 + n15];   b.y = Wk1[16 + n15];
        acc1 = __builtin_amdgcn_wmma_f32_16x16x4_f32(false, a, false, b, (short)0, acc1, false, false);
        b.x = Wk[32 + n15];   b.y = Wk1[32 + n15];
        acc2 = __builtin_amdgcn_wmma_f32_16x16x4_f32(false, a, false, b, (short)0, acc2, false, false);
        b.x = Wk[48 + n15];   b.y = Wk1[48 + n15];
        acc3 = __builtin_amdgcn_wmma_f32_16x16x4_f32(false, a, false, b, (short)0, acc3, false, false);
    }

    // D layout: VGPR v -> M = v (lanes 0-15) / v+8 (lanes 16-31), N = nt*16 + lane%16
    const float bias0 = b1[n15];
    const float bias1 = b1[16 + n15];
    const float bias2 = b1[32 + n15];
    const float bias3 = b1[48 + n15];
#pragma unroll
    for (int v = 0; v < 8; ++v) {
        const int m = (lane < 16) ? v : (v + 8);
        float* __restrict__ hrow = h + (long)(tile * 16 + m) * HID_CH;
        hrow[n15]      = fmaxf(acc0[v] + bias0, 0.0f);
        hrow[16 + n15] = fmaxf(acc1[v] + bias1, 0.0f);
        hrow[32 + n15] = fmaxf(acc2[v] + bias2, 0.0f);
        hrow[48 + n15] = fmaxf(acc3[v] + bias3, 0.0f);
    }
}

// ------------------------------------------------ layer-2 edge scatter (64 ch)
__global__ void sage_agg2_kernel(const float* __restrict__ h,
                                 const long long* __restrict__ src,
                                 const long long* __restrict__ dst,
                                 float* __restrict__ agg2) {
    long t = (long)blockIdx.x * blockDim.x + threadIdx.x;
    if (t >= (long)N_EDGES * HID_CH) return;
    int e = (int)(t >> 6);
    int c = (int)(t & 63);
    int s = (int)src[e];
    int d = (int)dst[e];
    atomicAdd(&agg2[(long)d * HID_CH + c], h[(long)s * HID_CH + c]);
}

// --------------------------------- layer-2: dot(64) + bias + sigmoid per node
__global__ void sage_l2_kernel(const float* __restrict__ agg2,
                               const float* __restrict__ deg,
                               const float* __restrict__ h,
                               const float* __restrict__ W2l,  // [64]
                               const float* __restrict__ W2r,  // [64]
                               const float* __restrict__ b2,   // [1]
                               float* __restrict__ out) {
    int i = blockIdx.x * blockDim.x + threadIdx.x;
    if (i >= N_NODES) return;
    const float inv = 1.0f / fmaxf(deg[i], 1.0f);
    const float* __restrict__ a  = agg2 + (long)i * HID_CH;
    const float* __restrict__ hh = h    + (long)i * HID_CH;
    float acc = b2[0];
#pragma unroll
    for (int k = 0; k < HID_CH; ++k) {
        acc += a[k] * inv * W2l[k] + hh[k] * W2r[k];
    }
    out[i] = 1.0f / (1.0f + expf(-acc));
}

// ---------------------------------------------------------------------- launch
extern "C" void kernel_launch(void* const* d_in, const int* in_sizes, int n_in,
                              void* d_out, int out_size, void* d_ws, size_t ws_size,
                              hipStream_t stream) {
    const float*     x   = (const float*)d_in[0];
    const long long* ei  = (const long long*)d_in[1];   // int64 [2, E]
    const float*     W1l = (const float*)d_in[2];
    const float*     W1r = (const float*)d_in[3];
    const float*     b1  = (const float*)d_in[4];
    const float*     W2l = (const float*)d_in[5];
    const float*     W2r = (const float*)d_in[6];
    const float*     b2  = (const float*)d_in[7];
    float*           out = (float*)d_out;

    const long long* src = ei;
    const long long* dst = ei + N_EDGES;

    // workspace layout (floats): agg1[N*32] | deg[N] | h[N*64] | agg2[N*64]
    float* agg1 = (float*)d_ws;
    float* deg  = agg1 + (size_t)N_NODES * IN_CH;
    float* h    = deg  + (size_t)N_NODES;
    float* agg2 = h    + (size_t)N_NODES * HID_CH;

    const int B = 256;

    // zero accumulators every call (harness does not re-poison between replays)
    {
        long n1 = (long)N_NODES * IN_CH + N_NODES;   // agg1 + deg (contiguous)
        sage_zero_kernel<<<(int)((n1 + B - 1) / B), B, 0, stream>>>(agg1, n1);
        long n2 = (long)N_NODES * HID_CH;            // agg2
        sage_zero_kernel<<<(int)((n2 + B - 1) / B), B, 0, stream>>>(agg2, n2);
    }

    // layer-1 aggregation
    {
        long n = (long)N_EDGES * IN_CH;
        sage_agg1_kernel<<<(int)((n + B - 1) / B), B, 0, stream>>>(x, src, dst, agg1, deg);
    }

    // layer-1 fused GEMM (fp32 WMMA): 6250 wave-tiles, 8 waves per 256-thread block
    {
        int tiles  = N_NODES / 16;                   // 6250 (N divisible by 16)
        int blocks = (tiles + 7) / 8;                // 782
        sage_l1_wmma_kernel<<<blocks, B, 0, stream>>>(agg1, deg, x, W1l, W1r, b1, h);
    }

    // layer-2 aggregation
    {
        long n = (long)N_EDGES * HID_CH;
        sage_agg2_kernel<<<(int)((n + B - 1) / B), B, 0, stream>>>(h, src, dst, agg2);
    }

    // layer-2 dot + sigmoid
    sage_l2_kernel<<<(N_NODES + B - 1) / B, B, 0, stream>>>(agg2, deg, h, W2l, W2r, b2, out);
}